// BSMatchStar_39298950758457
// MI455X (gfx1250) — compile-verified
//
#include <hip/hip_runtime.h>
#include <stdint.h>

// BSMatchStar on MI455X (gfx1250):
// single-pass decoupled-lookback carry scan over u64-packed words.
// Traffic: 512MB in + 256MB out = 768MB  ->  ~33us at 23.3 TB/s (roofline).
// Inputs staged into LDS via the Tensor Data Mover (tensor_load_to_lds),
// block scans use wave32 shuffle scans over the {kill,gen,prop} monoid.

#define BLOCK_THREADS 256
#define WPT 2                                   // u64 words per thread
#define TILE_WORDS (BLOCK_THREADS * WPT)        // 512 u64 words per block
#define TILE_INTS  (TILE_WORDS * 8)             // 4096 int32 per input array

typedef unsigned int u32;
typedef unsigned long long u64;
typedef u32 u32x4 __attribute__((ext_vector_type(4)));
typedef int i32x4 __attribute__((ext_vector_type(4)));
typedef int i32x8 __attribute__((ext_vector_type(8)));
typedef float f32x4 __attribute__((ext_vector_type(4)));   // true clang vector

// Flag monoid: 0=kill, 1=generate, 2=propagate.  op(l,r) = (r==prop) ? l : r.
// Identity = 2 (propagate).  Associative (carry-lookahead operator).
__device__ __forceinline__ u32 fop(u32 l, u32 r) { return (r == 2u) ? l : r; }

// Reverse bit order within each byte of a u64:
// bitreverse64 reverses bits AND bytes; bswap64 restores byte order.
__device__ __forceinline__ u64 brev_bytes(u64 x) {
  return __builtin_bswap64(__builtin_bitreverse64(x));
}

__device__ __forceinline__ u64 pack8(int4 v0, int4 v1) {
  u32 lo = (u32)(v0.x & 255) | ((u32)(v0.y & 255) << 8) |
           ((u32)(v0.z & 255) << 16) | ((u32)(v0.w & 255) << 24);
  u32 hi = (u32)(v1.x & 255) | ((u32)(v1.y & 255) << 8) |
           ((u32)(v1.z & 255) << 16) | ((u32)(v1.w & 255) << 24);
  return (u64)lo | ((u64)hi << 32);
}

// Issue one TDM DMA of TILE_INTS consecutive int32 (16KB) global -> LDS.
// D# group0: count=1 valid, lds_addr, 57-bit global addr, type=2 ("image").
// D# group1: data_size=2 (4B), tensor_dim0=tile_dim0=TILE_INTS (tile==tensor,
//            so no OOB), tensor_dim1=1, stride0=TILE_INTS.  No pad/iterate.
// NOTE: intentionally unconditional (no __has_builtin gate) so that if this
// toolchain lacks the builtin we get a hard diagnostic instead of silently
// compiling the non-TDM fallback.
__device__ __forceinline__ void tdm_load_tile(const int* gsrc, u32 lds_off) {
  u64 ga = (u64)(uintptr_t)gsrc;
  u32x4 g0;
  g0.x = 1u;                                        // count=1 (valid), user mode
  g0.y = lds_off;                                   // LDS byte address
  g0.z = (u32)ga;                                   // global_addr[31:0]
  g0.w = (u32)((ga >> 32) & 0x1FFFFFFu) | (2u << 30); // global_addr[56:32], type=2
  i32x8 g1;
  g1[0] = (int)(2u << 16);                          // data_size=2 (4 bytes)
  g1[1] = (int)((u32)TILE_INTS << 16);              // tensor_dim0[15:0]
  g1[2] = (int)(1u << 16);                          // tensor_dim0[31:16]=0, tensor_dim1=1
  g1[3] = (int)((u32)TILE_INTS << 16);              // tile_dim0
  g1[4] = 0;                                        // tile_dim1=0, tile_dim2=0 (1-D)
  g1[5] = (int)TILE_INTS;                           // tensor_dim0_stride[31:0]
  g1[6] = 0;
  g1[7] = 0;
  i32x4 gz = {0, 0, 0, 0};
#if __has_include(<hip/amd_detail/amd_gfx1250_TDM.h>)
  // amdgpu-toolchain (clang-23): 6-arg form
  i32x8 gz8 = {0, 0, 0, 0, 0, 0, 0, 0};
  __builtin_amdgcn_tensor_load_to_lds(g0, g1, gz, gz, gz8, 0);
#else
  // ROCm 7.2 (clang-22): 5-arg form
  __builtin_amdgcn_tensor_load_to_lds(g0, g1, gz, gz, 0);
#endif
}

__global__ void init_ws_kernel(u32* ws, u32 n) {
  u32 i = blockIdx.x * blockDim.x + threadIdx.x;
  if (i < n) ws[i] = 0u;
}

// status[] word: bits[1:0] = flag value {0,1,2}; bits[9:8] = state
// (0 = invalid, 1 = aggregate ready, 2 = inclusive prefix ready)
__global__ __launch_bounds__(BLOCK_THREADS) void bs_scan_kernel(
    const int* __restrict__ a, const int* __restrict__ b,
    float* __restrict__ out, u32* __restrict__ status,
    u32* __restrict__ counter) {
  __shared__ alignas(16) int sa[TILE_INTS];   // 16 KB
  __shared__ alignas(16) int sb[TILE_INTS];   // 16 KB
  __shared__ u32 s_wflags[BLOCK_THREADS / 32];
  __shared__ u32 s_bid;
  __shared__ u32 s_carry;

  const u32 tid  = threadIdx.x;
  const u32 lane = tid & 31u;
  const u32 wid  = tid >> 5;

  // Ticket: processing order == acquisition order => lookback cannot deadlock.
  if (tid == 0) s_bid = atomicAdd(counter, 1u);
  __syncthreads();
  const u32 bid = s_bid;

  const int* ga = a + (size_t)bid * TILE_INTS;
  const int* gb = b + (size_t)bid * TILE_INTS;

  // --- Tensor Data Mover: DMA both 16KB input tiles into LDS ---
  if (tid == 0) {
    tdm_load_tile(ga, (u32)(uintptr_t)(&sa[0]));
    tdm_load_tile(gb, (u32)(uintptr_t)(&sb[0]));
  }
  __builtin_amdgcn_s_wait_tensorcnt(0);   // issuing wave's DMA complete
  __syncthreads();                        // LDS visible to all 8 waves

  // ---- per-word: pack 8 bytes -> u64, reversed-domain add, gen/prop flags ----
  u64 sum[WPT], av[WPT], bv[WPT];
  u32 flg[WPT];
#pragma unroll
  for (int w = 0; w < WPT; ++w) {
    const int base4 = (int)((tid * WPT + (u32)w) * 8u);
    int4 a0 = *(const int4*)&sa[base4];
    int4 a1 = *(const int4*)&sa[base4 + 4];
    int4 b0 = *(const int4*)&sb[base4];
    int4 b1 = *(const int4*)&sb[base4 + 4];
    u64 a8 = pack8(a0, a1);
    u64 b8 = pack8(b0, b1);
    u64 ar = brev_bytes(a8 & b8);
    u64 br = brev_bytes(b8);
    u64 s  = ar + br;                       // 8 bytes of CLA add in one op
    flg[w] = (s < ar) ? 1u : ((s == ~0ull) ? 2u : 0u);
    sum[w] = s; av[w] = a8; bv[w] = b8;
  }
  u32 tf = flg[0];
#pragma unroll
  for (int w = 1; w < WPT; ++w) tf = fop(tf, flg[w]);

  // ---- wave32 shuffle scan of thread flags, then fold 8 wave aggregates ----
  u32 inc = tf;
#pragma unroll
  for (int d = 1; d < 32; d <<= 1) {
    u32 prev = __shfl_up(inc, d, 32);
    u32 comb = fop(prev, inc);
    inc = (lane >= (u32)d) ? comb : inc;
  }
  u32 laneExcl = __shfl_up(inc, 1, 32);
  if (lane == 0) laneExcl = 2u;
  if (lane == 31) s_wflags[wid] = inc;
  __syncthreads();
  u32 wexcl = 2u, blockIncl = 2u;
#pragma unroll
  for (u32 i = 0; i < BLOCK_THREADS / 32; ++i) {
    u32 wf = s_wflags[i];
    if (i < wid) wexcl = fop(wexcl, wf);
    blockIncl = fop(blockIncl, wf);
  }
  u32 threadExcl = fop(wexcl, laneExcl);

  // ---- decoupled lookback (wave 0): publish aggregate, fold predecessors ----
  if (wid == 0) {
    if (lane == 0) {
      u32 st = ((bid == 0) ? (2u << 8) : (1u << 8)) | blockIncl;
      __hip_atomic_store(&status[bid], st, __ATOMIC_RELEASE,
                         __HIP_MEMORY_SCOPE_AGENT);
    }
    u32 prefixFold = 2u;   // identity; initial stream carry-in is 0
    if (bid != 0) {
      int windowEnd = (int)bid - 1;
      u32 acc = 2u;
      for (;;) {
        int idx = windowEnd - 31 + (int)lane;   // lane 31 = nearest predecessor
        u32 st;
        if (idx >= 0) {
          for (;;) {
            st = __hip_atomic_load(&status[idx], __ATOMIC_ACQUIRE,
                                   __HIP_MEMORY_SCOPE_AGENT);
            if (((st >> 8) & 3u) != 0u) break;
            __builtin_amdgcn_s_sleep(1);
          }
        } else {
          st = (2u << 8) | 2u;                  // off the left edge: identity
        }
        u32 state = (st >> 8) & 3u;
        u32 contrib = st & 3u;
        u32 pmask = (u32)__ballot(state == 2u);
        if (pmask) {
          int hi = 31 - __builtin_clz(pmask);   // closest full prefix
          if ((int)lane < hi) contrib = 2u;     // identity left of it
        }
        u32 wf = contrib;                       // ordered fold across lanes
#pragma unroll
        for (int d = 1; d < 32; d <<= 1) {
          u32 prev = __shfl_up(wf, d, 32);
          u32 comb = fop(prev, wf);
          wf = (lane >= (u32)d) ? comb : wf;
        }
        u32 windowFold = __shfl(wf, 31, 32);
        acc = fop(windowFold, acc);
        if (pmask) { prefixFold = acc; break; }
        windowEnd -= 32;
      }
      if (lane == 0) {
        u32 inclP = fop(prefixFold, blockIncl);
        __hip_atomic_store(&status[bid], (2u << 8) | inclP, __ATOMIC_RELEASE,
                           __HIP_MEMORY_SCOPE_AGENT);
      }
    }
    if (lane == 0) s_carry = (prefixFold == 1u) ? 1u : 0u;
  }
  __syncthreads();
  const u32 carry = s_carry;

  // ---- resolve carries, finish combine, write 8 floats per word ----
  u32 c = (threadExcl == 1u) ? 1u : ((threadExcl == 2u) ? carry : 0u);
#pragma unroll
  for (int w = 0; w < WPT; ++w) {
    u64 res = sum[w] + (u64)c;
    u64 o = (brev_bytes(res) ^ bv[w]) | av[w];
    size_t wordIdx = (size_t)bid * TILE_WORDS + (size_t)(tid * WPT + (u32)w);
    float* op = out + wordIdx * 8;
    u32 olo = (u32)o, ohi = (u32)(o >> 32);
    f32x4 f0 = { (float)(olo & 255u), (float)((olo >> 8) & 255u),
                 (float)((olo >> 16) & 255u), (float)(olo >> 24) };
    f32x4 f1 = { (float)(ohi & 255u), (float)((ohi >> 8) & 255u),
                 (float)((ohi >> 16) & 255u), (float)(ohi >> 24) };
    __builtin_nontemporal_store(f0, (f32x4*)op);        // write-once stream
    __builtin_nontemporal_store(f1, (f32x4*)(op + 4));
    c = (flg[w] == 1u) ? 1u : ((flg[w] == 2u) ? c : 0u);
  }
}

extern "C" void kernel_launch(void* const* d_in, const int* in_sizes, int n_in,
                              void* d_out, int out_size, void* d_ws,
                              size_t ws_size, hipStream_t stream) {
  const int* a = (const int*)d_in[0];
  const int* b = (const int*)d_in[1];
  float* out = (float*)d_out;
  u32 nInts = (u32)in_sizes[0];          // 67,108,864 bytes (as int32 each)
  u32 nblocks = nInts / TILE_INTS;       // 16,384 blocks
  u32* wsw = (u32*)d_ws;
  u32* status = wsw;                     // nblocks status words
  u32* counter = wsw + nblocks;          // 1 ticket counter

  init_ws_kernel<<<(nblocks + 1 + 255) / 256, 256, 0, stream>>>(wsw, nblocks + 1);
  bs_scan_kernel<<<nblocks, BLOCK_THREADS, 0, stream>>>(a, b, out, status, counter);
}